// EmbeddingClassifier_49924699849369
// MI455X (gfx1250) — compile-verified
//
#include <hip/hip_runtime.h>
#include <cmath>

typedef float v2f __attribute__((ext_vector_type(2)));
typedef float v8f __attribute__((ext_vector_type(8)));

#define NE 11
#define ND 1536
#define NH 3072
#define NC 5242
#define NB 1024
#define BPAD 1280   // max compacted rows: 1024 + 11*15 padding, rounded up

// ---------------- workspace layout (bytes) ----------------
#define WS_COUNTS   0
#define WS_OFFSETS  64
#define WS_CURSOR   128
#define WS_BINS     192
#define WS_PERM     4352
#define WS_H1       9728
#define WS_H2       15738368

__global__ void route_init(int* counts, int* cursor, int* perm) {
    int i = blockIdx.x * blockDim.x + threadIdx.x;
    if (i < 16) { counts[i] = 0; cursor[i] = 0; }
    if (i < BPAD) perm[i] = -1;
}

__global__ void route_count(const float* __restrict__ mf, int* bins, int* counts) {
    int i = blockIdx.x * blockDim.x + threadIdx.x;
    if (i < NB) {
        float t = 1.0f - mf[i];
        int b = (int)(t / 0.1f);          // matches jnp f32 divide + int32 trunc
        b = b < 0 ? 0 : (b > NE - 1 ? NE - 1 : b);
        bins[i] = b;
        atomicAdd(&counts[b], 1);
    }
}

__global__ void route_offsets(const int* __restrict__ counts, int* offsets) {
    if (threadIdx.x == 0 && blockIdx.x == 0) {
        int run = 0;
        for (int e = 0; e < NE; ++e) {
            offsets[e] = run;
            run += (counts[e] + 15) & ~15;   // 16-align each expert's row range
        }
        offsets[NE] = run;
    }
}

__global__ void route_scatter(const int* __restrict__ bins,
                              const int* __restrict__ offsets,
                              int* cursor, int* perm) {
    int i = blockIdx.x * blockDim.x + threadIdx.x;
    if (i < NB) {
        int e = bins[i];
        int p = atomicAdd(&cursor[e], 1);
        perm[offsets[e] + p] = i;
    }
}

// Grouped-expert GEMM: block tile 64(M) x 128(N), 8 waves (wm 0..3, wn 0..1),
// each wave 16x64 via 4 f32 WMMA accumulators, K chunked by 32 through a
// double-buffered LDS tile (global loads for chunk c+1 overlap WMMA on chunk c).
template <bool GATHER_IN, bool DO_GELU, bool SCATTER_OUT>
__global__ __launch_bounds__(256)
void expert_gemm(const float* __restrict__ A, const float* __restrict__ Wall,
                 const float* __restrict__ Ball, float* __restrict__ Out,
                 const int* __restrict__ perm, const int* __restrict__ offsets,
                 const int* __restrict__ counts,
                 int K, int N, int ldA, int ldOut)
{
    const int e      = blockIdx.z;
    const int roff   = offsets[e];
    const int cnt    = counts[e];
    const int cntPad = offsets[e + 1] - roff;
    const int mBase  = blockIdx.y * 64;
    if (mBase >= cntPad) return;                 // dynamic early-exit per expert
    const int n0 = blockIdx.x * 128;

    const float* __restrict__ W    = Wall + (size_t)e * (size_t)K * (size_t)N;
    const float* __restrict__ bias = Ball + (size_t)e * (size_t)N;

    const int tid  = threadIdx.x;
    const int lane = tid & 31;
    const int w    = tid >> 5;
    const int wm   = w & 3;        // M sub-tile 0..3
    const int wn   = w >> 2;       // N half 0..1

    // A fragment addressing: lane&15 = M row, K = 2*(lane>>4) + elem
    const int rA = roff + mBase + wm * 16 + (lane & 15);
    int srcRow;
    if (GATHER_IN) { int p = perm[rA]; srcRow = (p < 0) ? 0 : p; }
    else           { srcRow = rA; }
    const float* __restrict__ aRow = A + (size_t)srcRow * (size_t)ldA + 2 * (lane >> 4);

    // B fragments pre-permuted: [buf][nsub][kstep][lane][elem]; lane reads its
    // own 8 bytes -> conflict-free ds_load_2addr_b64 pairs. Double buffered.
    __shared__ float Bs[2][8][8][32][2];         // 32 KB

    // per-thread staging indices for the 32x128 W chunk (float2 granularity)
    int skk[8], snn2[8];
    #pragma unroll
    for (int i = 0; i < 8; ++i) {
        int flat2 = tid + i * 256;               // 0..2047 float2 slots
        skk[i]  = flat2 >> 6;                    // 0..31
        snn2[i] = (flat2 & 63) << 1;             // 0..126 (even)
    }

    float r0[8], r1[8];

    // ---- chunk loader into registers (N always even; handles NC=5242 tail)
    auto loadW = [&](int kB) {
        #pragma unroll
        for (int i = 0; i < 8; ++i) {
            int n = n0 + snn2[i];
            float a0 = 0.0f, a1 = 0.0f;
            const float* src = W + (size_t)(kB + skk[i]) * (size_t)N + n;
            if (n + 1 < N) { v2f v = *(const v2f*)src; a0 = v.x; a1 = v.y; }
            else if (n < N) { a0 = *src; }
            r0[i] = a0; r1[i] = a1;
        }
    };
    // ---- registers -> permuted LDS fragments
    auto storeW = [&](int buf) {
        #pragma unroll
        for (int i = 0; i < 8; ++i) {
            int kk = skk[i], nn2 = snn2[i];
            int kstep = kk >> 2;
            int elem  = kk & 1;
            int lh    = ((kk & 3) >> 1) << 4;    // lane-half from K pair
            int nsub  = nn2 >> 4;
            int lf    = (nn2 & 15) + lh;
            Bs[buf][nsub][kstep][lf][elem]     = r0[i];
            Bs[buf][nsub][kstep][lf + 1][elem] = r1[i];
        }
    };

    v8f acc0 = {}, acc1 = {}, acc2 = {}, acc3 = {};

    const int nChunks = K >> 5;
    loadW(0);
    storeW(0);
    __syncthreads();

    // L2 prefetch geometry: 256 threads cover the 32x128 chunk
    const int pfRow = tid >> 3;                  // 0..31
    int pfCol = n0 + (tid & 7) * 16;             // 64B strided
    if (pfCol > N - 1) pfCol = N - 1;            // never beyond the W allocation

    for (int c = 0; c < nChunks; ++c) {
        const int k0  = c << 5;
        const int buf = c & 1;

        if (c + 1 < nChunks) {
            loadW(k0 + 32);                       // global loads fly during WMMAs
            if (c + 2 < nChunks)
                __builtin_prefetch(W + (size_t)(k0 + 64 + pfRow) * (size_t)N + pfCol, 0, 1);
        }

        #pragma unroll
        for (int ks = 0; ks < 8; ++ks) {
            v2f af = *(const v2f*)(aRow + k0 + 4 * ks);
            v2f b0 = *(const v2f*)&Bs[buf][wn * 4 + 0][ks][lane][0];
            v2f b1 = *(const v2f*)&Bs[buf][wn * 4 + 1][ks][lane][0];
            v2f b2 = *(const v2f*)&Bs[buf][wn * 4 + 2][ks][lane][0];
            v2f b3 = *(const v2f*)&Bs[buf][wn * 4 + 3][ks][lane][0];
            acc0 = __builtin_amdgcn_wmma_f32_16x16x4_f32(false, af, false, b0, (short)0, acc0, false, false);
            acc1 = __builtin_amdgcn_wmma_f32_16x16x4_f32(false, af, false, b1, (short)0, acc1, false, false);
            acc2 = __builtin_amdgcn_wmma_f32_16x16x4_f32(false, af, false, b2, (short)0, acc2, false, false);
            acc3 = __builtin_amdgcn_wmma_f32_16x16x4_f32(false, af, false, b3, (short)0, acc3, false, false);
        }

        if (c + 1 < nChunks) storeW(buf ^ 1);
        __syncthreads();
    }

    // epilogue: C/D layout lane&15 = N, M = vg + 8*(lane>>4)
    const int nLane = lane & 15;
    const int mHalf = 8 * (lane >> 4);
    v8f accs[4] = {acc0, acc1, acc2, acc3};
    #pragma unroll
    for (int t = 0; t < 4; ++t) {
        int n = n0 + wn * 64 + t * 16 + nLane;
        if (n >= N) continue;
        float bv = bias[n];
        #pragma unroll
        for (int vg = 0; vg < 8; ++vg) {
            int mLocal = mBase + wm * 16 + mHalf + vg;
            float val = accs[t][vg] + bv;
            if (DO_GELU)
                val = 0.5f * val * (1.0f + erff(val * 0.7071067811865475f));
            if (SCATTER_OUT) {
                if (mLocal < cnt) {
                    int s = perm[roff + mLocal];
                    if (s >= 0) Out[(size_t)s * (size_t)ldOut + n] = val;
                }
            } else {
                if (mLocal < cntPad)   // never cross into next expert's rows
                    Out[(size_t)(roff + mLocal) * (size_t)ldOut + n] = val;
            }
        }
    }
}

__global__ void logsoftmax_rows(float* __restrict__ out) {
    __shared__ float red[256];
    float* p = out + (size_t)blockIdx.x * NC;
    int t = threadIdx.x;

    float mx = -INFINITY;
    for (int c = t; c < NC; c += 256) mx = fmaxf(mx, p[c]);
    red[t] = mx; __syncthreads();
    for (int s = 128; s > 0; s >>= 1) {
        if (t < s) red[t] = fmaxf(red[t], red[t + s]);
        __syncthreads();
    }
    mx = red[0]; __syncthreads();

    float sum = 0.0f;
    for (int c = t; c < NC; c += 256) sum += expf(p[c] - mx);
    red[t] = sum; __syncthreads();
    for (int s = 128; s > 0; s >>= 1) {
        if (t < s) red[t] += red[t + s];
        __syncthreads();
    }
    float lse = logf(red[0]);
    for (int c = t; c < NC; c += 256) p[c] = p[c] - mx - lse;
}

extern "C" void kernel_launch(void* const* d_in, const int* in_sizes, int n_in,
                              void* d_out, int out_size, void* d_ws, size_t ws_size,
                              hipStream_t stream) {
    const float* x  = (const float*)d_in[0];
    const float* mf = (const float*)d_in[1];
    const float* W1 = (const float*)d_in[2];
    const float* b1 = (const float*)d_in[3];
    const float* W2 = (const float*)d_in[4];
    const float* b2 = (const float*)d_in[5];
    const float* W3 = (const float*)d_in[6];
    const float* b3 = (const float*)d_in[7];
    float* out = (float*)d_out;

    char* ws = (char*)d_ws;
    int*   counts  = (int*)(ws + WS_COUNTS);
    int*   offsets = (int*)(ws + WS_OFFSETS);
    int*   cursor  = (int*)(ws + WS_CURSOR);
    int*   bins    = (int*)(ws + WS_BINS);
    int*   perm    = (int*)(ws + WS_PERM);
    float* h1      = (float*)(ws + WS_H1);
    float* h2      = (float*)(ws + WS_H2);

    route_init   <<<(BPAD + 255) / 256, 256, 0, stream>>>(counts, cursor, perm);
    route_count  <<<(NB + 255) / 256, 256, 0, stream>>>(mf, bins, counts);
    route_offsets<<<1, 32, 0, stream>>>(counts, offsets);
    route_scatter<<<(NB + 255) / 256, 256, 0, stream>>>(bins, offsets, cursor, perm);

    // layer 1: [cnt,1536] x [1536,3072] + b1, GELU -> h1 (compacted)
    expert_gemm<true, true, false><<<dim3(NH / 128, 16, NE), 256, 0, stream>>>(
        x, W1, b1, h1, perm, offsets, counts, ND, NH, ND, NH);
    // layer 2: [cnt,3072] x [3072,3072] + b2, GELU -> h2 (compacted)
    expert_gemm<false, true, false><<<dim3(NH / 128, 16, NE), 256, 0, stream>>>(
        h1, W2, b2, h2, perm, offsets, counts, NH, NH, NH, NH);
    // layer 3: [cnt,3072] x [3072,5242] + b3 -> scatter logits to out[sample]
    expert_gemm<false, false, true><<<dim3((NC + 127) / 128, 16, NE), 256, 0, stream>>>(
        h2, W3, b3, out, perm, offsets, counts, NH, NC, NH, NC);

    logsoftmax_rows<<<NB, 256, 0, stream>>>(out);
}